// Primitive3DRenderer_4440996184875
// MI455X (gfx1250) — compile-verified
//
#include <hip/hip_runtime.h>
#include <math.h>

typedef __attribute__((ext_vector_type(2))) float v2f;
typedef __attribute__((ext_vector_type(8))) float v8f;

#define RES    32
#define NVOX   (RES * RES * RES)   /* 32768 voxels per batch */
#define NPRIM  64

// Integer max (valid float ordering for non-negative floats; all candidates
// here are >= 0). Avoids llvm.maxnum's sNaN canonicalization copies and can
// use v_max_i32 / v_max3_i32.
__device__ __forceinline__ int imax2(int a, int b) { return (a > b) ? a : b; }

// DPP-based lane-permute max in the integer domain: ideally a single
// v_max_i32 with a DPP source. CTRL: 0xB1 quad_perm xor1, 0x4E quad_perm
// xor2, 0x141 row_half_mirror (xor4 once quads uniform), 0x140 row_mirror
// (xor8 once octs uniform). All stay within a 16-lane row (wave32 halves).
template <int CTRL>
__device__ __forceinline__ int dpp_imax(int m) {
    int p = __builtin_amdgcn_update_dpp(0, m, CTRL, 0xf, 0xf, true);
    return imax2(m, p);
}

// One block = 256 threads = 8 wave32s. Each wave processes 8 groups of 16
// voxels (1024 voxels / block); 32 blocks per batch, 64 batches in grid.y.
// dist^2 < r^2 is tested as (|v|^2 - 2 v.c) < (r^2 - |c|^2), with the LHS
// produced as a 16x16 tile by V_WMMA_F32_16X16X4_F32 (A = voxels x [x,y,z,|v|^2],
// B = primitives x [-2cx,-2cy,-2cz,1]).
__global__ __launch_bounds__(256) void prim3d_wmma_kernel(
    const float* __restrict__ z_t,
    const float* __restrict__ z_b_shared,
    const float* __restrict__ z_b_style,
    const float* __restrict__ coords,
    float* __restrict__ out)
{
    const int batch = blockIdx.y;

    __shared__ float sVal[NPRIM];     // intensity * exists (>= 0)
    __shared__ float sThresh[NPRIM];  // r^2 - |c|^2

    const int tid = threadIdx.x;

    // --- per-batch primitive preprocessing (64 threads) -------------------
    if (tid < NPRIM) {
        float zt = z_t[batch * NPRIM + tid];
        float r  = 0.12f + zt * 0.04f;                  // 0.12 + zt/25
        float vl = (z_b_shared[batch * NPRIM + tid] +
                    z_b_style[batch * NPRIM + tid]) * 0.03125f;  // /32
        vl = (zt > 0.0f) ? vl : 0.0f;                   // mask_exists
        int pi = tid >> 4, pj = (tid >> 2) & 3, pk = tid & 3;
        float cx = ((float)pi - 1.5f) * 0.5f;
        float cy = ((float)pj - 1.5f) * 0.5f;
        float cz = ((float)pk - 1.5f) * 0.5f;
        float c2 = cx * cx + cy * cy + cz * cz;
        sVal[tid]    = vl;
        sThresh[tid] = r * r - c2;     // inside  <=>  (|v|^2 - 2 v.c) < thresh
    }
    __syncthreads();

    const int  lane = tid & 31;
    const int  wave = tid >> 5;
    const int  col  = lane & 15;       // D-matrix column (primitive within tile)
    const bool hi   = lane >= 16;      // lane half: rows 8..15 / K=2,3 slots

    // --- per-lane, per-tile constants: B-matrix entries, thresh, val bits -
    float b0[4], b1[4], th[4];
    int   vb[4];                       // bit patterns of sVal (non-negative)
#pragma unroll
    for (int t = 0; t < 4; ++t) {
        int p  = t * 16 + col;
        int pi = p >> 4, pj = (p >> 2) & 3, pk = p & 3;
        float cx = ((float)pi - 1.5f) * 0.5f;
        float cy = ((float)pj - 1.5f) * 0.5f;
        float cz = ((float)pk - 1.5f) * 0.5f;
        // B 4x16 layout: lanes 0-15 hold K=0,1 rows; lanes 16-31 hold K=2,3.
        b0[t] = hi ? (-2.0f * cz) : (-2.0f * cx);
        b1[t] = hi ? 1.0f         : (-2.0f * cy);
        th[t] = sThresh[p];
        vb[t] = __float_as_int(sVal[p]);
    }

    const v8f czero = {0.f, 0.f, 0.f, 0.f, 0.f, 0.f, 0.f, 0.f};
    const int blockBase = blockIdx.x * 1024;
    const int waveBase  = blockBase + wave * 128;      // first voxel of this wave
    float* obase = out + (size_t)batch * NVOX;

    // This lane's voxel xyz for group g lives at cptr + g*48 (16 voxels * 3 floats).
    const float* cptr = coords + (size_t)(waveBase + col) * 3;

    // ---- software pipeline: branch-free prefetch with a clamped pointer.
    // cnext walks groups 1..7 and then sticks at 7 (last prefetch is a
    // harmless in-bounds re-read whose value is discarded). ----------------
    float x = cptr[0];
    float y = cptr[1];
    float z = cptr[2];
    const float* cnext = cptr + 48;

#pragma unroll 2
    for (int g = 0; g < 8; ++g) {
        // issue next group's load before doing any compute on this group
        float xn = cnext[0];
        float yn = cnext[1];
        float zn = cnext[2];

        // A 16x4 layout: lanes 0-15 hold (K0,K1)=(x,y); lanes 16-31 (K2,K3)=(z,|v|^2)
        float s = x * x + y * y + z * z;
        v2f A;
        A.x = hi ? z : x;
        A.y = hi ? s : y;

        int mx[8];

#pragma unroll
        for (int t = 0; t < 4; ++t) {
            v2f B;
            B.x = b0[t];
            B.y = b1[t];
            // D[m,n] = |v_m|^2 - 2 v_m . c_n   (16 voxels x 16 primitives)
            v8f D = __builtin_amdgcn_wmma_f32_16x16x4_f32(
                false, A, false, B, (short)0, czero, false, false);
#pragma unroll
            for (int e = 0; e < 8; ++e) {
                int sel = (D[e] < th[t]) ? vb[t] : 0;    // v_cmp_lt_f32 + v_cndmask
                if (t == 0) mx[e] = sel;                 // first tile: no max
                else        mx[e] = imax2(mx[e], sel);   // v_max_i32 / v_max3_i32
            }
        }

        // max over primitives = max over 16 lanes of each half (DPP butterfly)
#pragma unroll
        for (int e = 0; e < 8; ++e) {
            int m = mx[e];
            m = dpp_imax<0xB1>(m);    // quad_perm [1,0,3,2]  (xor 1)
            m = dpp_imax<0x4E>(m);    // quad_perm [2,3,0,1]  (xor 2)
            m = dpp_imax<0x141>(m);   // row_half_mirror      (xor 4 eq.)
            m = dpp_imax<0x140>(m);   // row_mirror           (xor 8 eq.)
            mx[e] = m;
        }

        // lane 0 holds voxels base..base+7 (rows 0-7), lane 16 holds base+8..15
        if (col == 0) {
            const int rowBase = waveBase + g * 16 + (hi ? 8 : 0);
            float m0 = __int_as_float(mx[0]);
            float m1 = __int_as_float(mx[1]);
            float m2 = __int_as_float(mx[2]);
            float m3 = __int_as_float(mx[3]);
            float m4 = __int_as_float(mx[4]);
            float m5 = __int_as_float(mx[5]);
            float m6 = __int_as_float(mx[6]);
            float m7 = __int_as_float(mx[7]);
            float4 o0, o1;
            o0.x = m0 * sqrtf(m0);
            o0.y = m1 * sqrtf(m1);
            o0.z = m2 * sqrtf(m2);
            o0.w = m3 * sqrtf(m3);
            o1.x = m4 * sqrtf(m4);
            o1.y = m5 * sqrtf(m5);
            o1.z = m6 * sqrtf(m6);
            o1.w = m7 * sqrtf(m7);
            *(float4*)(obase + rowBase)     = o0;
            *(float4*)(obase + rowBase + 4) = o1;
        }

        x = xn;
        y = yn;
        z = zn;
        // advance prefetch pointer; stick at group 7 (stays in bounds)
        cnext += (g < 6) ? 48 : 0;
    }
}

extern "C" void kernel_launch(void* const* d_in, const int* in_sizes, int n_in,
                              void* d_out, int out_size, void* d_ws, size_t ws_size,
                              hipStream_t stream) {
    (void)in_sizes; (void)n_in; (void)d_ws; (void)ws_size; (void)out_size;
    const float* z_t    = (const float*)d_in[0];
    const float* z_b_sh = (const float*)d_in[1];
    const float* z_b_st = (const float*)d_in[2];
    const float* coords = (const float*)d_in[3];
    float* out = (float*)d_out;

    dim3 grid(32, 64);   // 32 blocks x 1024 voxels per batch, 64 batches
    dim3 block(256);
    prim3d_wmma_kernel<<<grid, block, 0, stream>>>(z_t, z_b_sh, z_b_st, coords, out);
}